// TotalLoss_6236292513944
// MI455X (gfx1250) — compile-verified
//
#include <hip/hip_runtime.h>
#include <math.h>

#define WIDTH  1280
#define HEIGHT 384
#define NPTS   131072
#define NB     16

typedef float v2f __attribute__((ext_vector_type(2)));
typedef float v8f __attribute__((ext_vector_type(8)));

// workspace layout (floats): [0]=pc err accum, [1]=depth accum, predDM at offset 256
#define ACC_PC      0
#define ACC_DEPTH   1
#define PRED_DM_OFF 256

__device__ __forceinline__ float fast_sqrt(float x) { return __builtin_amdgcn_sqrtf(x); }
__device__ __forceinline__ float fast_rcp(float x)  { return __builtin_amdgcn_rcpf(x); }

__device__ __forceinline__ void rodrigues_rt(const float r0, const float r1, const float r2,
                                             const float t0, const float t1, const float t2,
                                             float rt[16]) {
    float th2 = r0 * r0 + r1 * r1 + r2 * r2;
    float th  = fast_sqrt(th2);
    float s   = sinf(th), c = cosf(th);
    float A   = s / th;
    float B   = (1.0f - c) / th2;
    rt[0]  = c + B * r0 * r0;        rt[1]  = -A * r2 + B * r0 * r1;  rt[2]  =  A * r1 + B * r0 * r2;  rt[3]  = t0;
    rt[4]  =  A * r2 + B * r0 * r1;  rt[5]  = c + B * r1 * r1;        rt[6]  = -A * r0 + B * r1 * r2;  rt[7]  = t1;
    rt[8]  = -A * r1 + B * r0 * r2;  rt[9]  =  A * r0 + B * r1 * r2;  rt[10] = c + B * r2 * r2;        rt[11] = t2;
    rt[12] = 0.f; rt[13] = 0.f; rt[14] = 0.f; rt[15] = 1.f;
}

// One block = (chunk of 2048 points, sample b). 8 waves, 256 points/wave, 32 points/iter.
__global__ __launch_bounds__(256)
void point_pass(const float* __restrict__ pc,     // (B,1,N,4)
                const float* __restrict__ pr_t,   // (B,3)
                const float* __restrict__ pr_r,   // (B,3)
                const float* __restrict__ gt_rt,  // (B,4,4)
                const float* __restrict__ Km,     // (3,3)
                float* __restrict__ ws)
{
    __shared__ float sMdiff[16];
    const int b    = blockIdx.y;
    const int tid  = threadIdx.x;
    const int lane = tid & 31;
    const int wave = tid >> 5;

    float rt[16];
    rodrigues_rt(pr_r[b * 3 + 0], pr_r[b * 3 + 1], pr_r[b * 3 + 2],
                 pr_t[b * 3 + 0], pr_t[b * 3 + 1], pr_t[b * 3 + 2], rt);

    // KR = K @ rt[:3,:]  (3x4), uniform in registers
    float KR[12];
#pragma unroll
    for (int i = 0; i < 3; ++i)
#pragma unroll
        for (int j = 0; j < 4; ++j)
            KR[i * 4 + j] = Km[i * 3 + 0] * rt[j] + Km[i * 3 + 1] * rt[4 + j] + Km[i * 3 + 2] * rt[8 + j];

    if (tid == 0) {
#pragma unroll
        for (int k = 0; k < 16; ++k) sMdiff[k] = rt[k] - gt_rt[b * 16 + k];
    }
    __syncthreads();

    // A fragment of V_WMMA_F32_16X16X4_F32: A = Mdiff, rows 0-3 valid, rows 4-15 zero.
    // A layout: lanes 0-15 hold (M=lane, K=0,1); lanes 16-31 hold (M=lane-16, K=2,3).
    const int m = lane & 15;
    const int h = lane >> 4;
    v2f afrag; afrag.x = 0.f; afrag.y = 0.f;
    if (m < 4) { afrag.x = sMdiff[m * 4 + 2 * h]; afrag.y = sMdiff[m * 4 + 2 * h + 1]; }

    const float4* pc4    = (const float4*)(pc) + (size_t)b * NPTS;
    float*        predDM = ws + PRED_DM_OFF + (size_t)b * (HEIGHT * WIDTH);

    const int base = (blockIdx.x * 8 + wave) * 256;
    float acc = 0.f;

    for (int it = 0; it < 8; ++it) {
        const int p0 = base + it * 32;
        float4 p = pc4[p0 + lane];

        // projection + depth scatter (one point per lane)
        float px = KR[0] * p.x + KR[1] * p.y + KR[2]  * p.z + KR[3]  * p.w;
        float py = KR[4] * p.x + KR[5] * p.y + KR[6]  * p.z + KR[7]  * p.w;
        float Z  = KR[8] * p.x + KR[9] * p.y + KR[10] * p.z + KR[11] * p.w;
        if (Z > 0.f) {
            float rz = fast_rcp(Z);
            float x = fminf(fmaxf(px * rz, 0.f), (float)(WIDTH - 1));
            float y = fminf(fmaxf(py * rz, 0.f), (float)(HEIGHT - 1));
            int xi = (int)x, yi = (int)y;
            predDM[yi * WIDTH + xi] = Z;   // .at[].set semantics: unspecified winner
        }

        // per-point L2 error via WMMA: D = Mdiff(16x4) @ pts^T(4x16).
        // D layout: VGPR v, lanes 0-15 -> (M=v, N=lane); so point N's diff coords
        // dx,dy,dz,dw sit in THE SAME lane across d[0..3]  ->  in-lane err^2, no shuffles.
        // Lanes 16-31 hold rows M=8..15 (zero) -> contribute sqrt(0)=0.
#pragma unroll
        for (int hh = 0; hh < 2; ++hh) {
            // B fragment: lane holds coords (2h, 2h+1) of point N=m  (cache-hot b64 load)
            const float* bp = (const float*)(pc4 + (p0 + hh * 16 + m)) + 2 * h;
            v2f bv; bv.x = bp[0]; bv.y = bp[1];
            v8f c = {0.f, 0.f, 0.f, 0.f, 0.f, 0.f, 0.f, 0.f};
            v8f d = __builtin_amdgcn_wmma_f32_16x16x4_f32(false, afrag, false, bv,
                                                          (short)0, c, false, false);
            float e2 = d[0] * d[0] + d[1] * d[1] + d[2] * d[2] + d[3] * d[3];
            acc += fast_sqrt(e2);
        }
    }

    // wave reduce (lanes 0-15 carry real errs, 16-31 carry zeros)
    acc += __shfl_xor(acc, 16, 32);
    acc += __shfl_xor(acc, 8, 32);
    acc += __shfl_xor(acc, 4, 32);
    acc += __shfl_xor(acc, 2, 32);
    acc += __shfl_xor(acc, 1, 32);
    if (lane == 0) atomicAdd(&ws[ACC_PC], acc);
}

// One block = (64-column tile, sample b); 4 row-groups of 96 rows each.
__global__ __launch_bounds__(256)
void depth_pass(const float* __restrict__ gt_dm, float* __restrict__ ws)
{
    __shared__ float red[256];
    const int b   = blockIdx.y;
    const int tid = threadIdx.x;
    const int col = blockIdx.x * 64 + (tid & 63);
    const int rg  = tid >> 6;

    const float* g = gt_dm + ((size_t)b * HEIGHT + rg * 96) * WIDTH + col;
    const float* p = ws + PRED_DM_OFF + ((size_t)b * HEIGHT + rg * 96) * WIDTH + col;

    float s = 0.f;
#pragma unroll 4
    for (int r = 0; r < 96; ++r) {
        float d = p[r * WIDTH] - g[r * WIDTH];
        s += d * d;
    }
    red[tid] = s;
    __syncthreads();

    float e = 0.f;
    if (tid < 64) {
        float tot = red[tid] + red[tid + 64] + red[tid + 128] + red[tid + 192];
        e = fast_sqrt(tot);   // per-column L2 over 384 rows
    }
    __syncthreads();
    red[tid] = e;
    __syncthreads();

    if (tid < 32) {
        float v = red[tid] + red[tid + 32];
        v += __shfl_xor(v, 16, 32);
        v += __shfl_xor(v, 8, 32);
        v += __shfl_xor(v, 4, 32);
        v += __shfl_xor(v, 2, 32);
        v += __shfl_xor(v, 1, 32);
        if (tid == 0) atomicAdd(&ws[ACC_DEPTH], v);
    }
}

__global__ void finalize(const float* __restrict__ gt_t, const float* __restrict__ gt_r,
                         const float* __restrict__ pr_t, const float* __restrict__ pr_r,
                         const float* __restrict__ ws, float* __restrict__ out)
{
    if (threadIdx.x != 0) return;
    float lt = 0.f, lr = 0.f;
    for (int b = 0; b < NB; ++b) {
        for (int j = 0; j < 3; ++j) {
            float dt = pr_t[b * 3 + j] - gt_t[b * 3 + j];
            float dr = pr_r[b * 3 + j] - gt_r[b * 3 + j];
            lt += dt * dt;
            lr += dr * dr;
        }
    }
    lt /= (float)NB; lr /= (float)NB;
    float trans = lt + 2.0f * lr;
    float pc_loss = ws[ACC_PC] / ((float)NPTS * (float)NB);
    float depth   = ws[ACC_DEPTH] / ((float)WIDTH * (float)NB);
    out[0] = 4.0f * trans + 1.0f * depth + 40.0f * pc_loss;
    out[1] = trans;
    out[2] = depth;
    out[3] = pc_loss;
}

extern "C" void kernel_launch(void* const* d_in, const int* in_sizes, int n_in,
                              void* d_out, int out_size, void* d_ws, size_t ws_size,
                              hipStream_t stream) {
    const float* pc    = (const float*)d_in[0];
    const float* gt_t  = (const float*)d_in[1];
    const float* gt_r  = (const float*)d_in[2];
    const float* pr_t  = (const float*)d_in[3];
    const float* pr_r  = (const float*)d_in[4];
    const float* gt_rt = (const float*)d_in[5];
    const float* Km    = (const float*)d_in[6];
    const float* gt_dm = (const float*)d_in[7];
    float* out = (float*)d_out;
    float* ws  = (float*)d_ws;

    const size_t clear_bytes = (PRED_DM_OFF + (size_t)NB * HEIGHT * WIDTH) * sizeof(float);
    hipMemsetAsync(d_ws, 0, clear_bytes, stream);

    dim3 gP(NPTS / 2048, NB);      // 64 x 16 blocks
    point_pass<<<gP, 256, 0, stream>>>(pc, pr_t, pr_r, gt_rt, Km, ws);

    dim3 gD(WIDTH / 64, NB);       // 20 x 16 blocks
    depth_pass<<<gD, 256, 0, stream>>>(gt_dm, ws);

    finalize<<<1, 32, 0, stream>>>(gt_t, gt_r, pr_t, pr_r, ws, out);
}